// Point_contextual_hyper_prior_encoder_6047313953100
// MI455X (gfx1250) — compile-verified
//
#include <hip/hip_runtime.h>

// ---------------------------------------------------------------------------
// Fused sparse-conv layer for CDNA5 (gfx1250), atomic-free, pipelined.
//
// Rulebook property (from reference): rb_out[k][m] == m on hit, sentinel on
// miss, with the miss mask identical to rb_in's.  Missed taps gather the zero
// row.  So:  out[m] = sum_k (rb_in[k][m] < n_in ? feats[rb_in[k][m]] : 0) @ W_k + b
// -> accumulate over k in registers, store once, fuse bias+ReLU.  No atomics,
// no output zero-init.  Remaining traffic = gathers (~1.24 GB, ~55us floor at
// 23.3 TB/s), so fp32 V_WMMA_F32_16X16X4_F32 keeps reference precision at no
// roofline cost.
//
// Pipelining:
//  * LDS weight tile double-buffered (2 x 16KB): stage W_{k+1} right after the
//    barrier, compute k from the other buffer -> global->LDS latency overlaps
//    compute; b128 staging reads.
//  * A-fragments prefetched one K-step ahead in distinct registers so the
//    compiler can issue loads early with staggered loadcnt waits.
//  * rb indices for k+1 prefetched during k's compute.
//
// Fragment layouts per ISA 7.12.2 (16x16x4 f32):
//   A 16x4 : 2 VGPRs, lanes 0-15 K={0,1}, lanes 16-31 K={2,3}
//   B 4x16 : mirrored K striping -> pair-interleaved LDS layout
//            lds[((r>>1)*64 + c)*2 + (r&1)] makes a fragment one b64 load
//            (and ds_load_2addr_b64 fetches two N-blocks at once).
//   C 16x16: 8 VGPRs, VGPR v <-> row v + 8*(lane>=16), col = lane&15
// ---------------------------------------------------------------------------

typedef __attribute__((ext_vector_type(2))) float v2f;
typedef __attribute__((ext_vector_type(8))) float v8f;

template <int RELU>
__global__ __launch_bounds__(256) void sparse_conv_fused(
    const float* __restrict__ feats,   // [n_in, 64]
    const float* __restrict__ W,       // [K, 64, 64]
    const float* __restrict__ bias,    // [64]
    const int*   __restrict__ rb_in,   // [K, M] (sentinel = n_in)
    float*       __restrict__ out,     // [M, 64] written exactly once
    int K, int n_in, int M, int ngroups)
{
  __shared__ float lds_w[2][64 * 64];  // double-buffered, pair-interleaved

  const int tid  = threadIdx.x;
  const int lane = tid & 31;
  const int hi   = lane >> 4;    // half-wave: K rows {0,1} vs {2,3}
  const int l15  = lane & 15;
  const int wv   = tid >> 5;     // wave id in block (0..7)

  // Per-lane bias fragment: column nb*16 + l15 (layer-constant).
  float bfr[4];
#pragma unroll
  for (int nb = 0; nb < 4; ++nb) bfr[nb] = bias[nb * 16 + l15];

  for (int tg = blockIdx.x; tg < ngroups; tg += gridDim.x) {
    const int tile0 = (tg * 8 + wv) * 2;   // two consecutive 16-row tiles
    const int row0  = tile0 * 16 + l15;
    const int row1  = row0 + 16;

    v8f acc[2][4];
#pragma unroll
    for (int t = 0; t < 2; ++t)
#pragma unroll
      for (int nb = 0; nb < 4; ++nb) acc[t][nb] = (v8f){};

    // Prefetch rulebook indices for k = 0.
    int ni0 = (row0 < M) ? rb_in[row0] : n_in;
    int ni1 = (row1 < M) ? rb_in[row1] : n_in;

    // Stage W_0 into buffer 0 (barrier protects previous tg's readers).
    __syncthreads();
    {
      const float4* Wk4 = (const float4*)W;   // k = 0
      float* lwb = lds_w[0];
#pragma unroll
      for (int t = 0; t < 4; ++t) {
        const int i4 = tid + t * 256;        // float4 index 0..1023
        const float4 w = Wk4[i4];
        const int r  = i4 >> 4;              // row 0..63
        const int c0 = (i4 & 15) * 4;        // col 0,4,...,60
        const int d  = (((r >> 1) * 64 + c0) << 1) + (r & 1);
        lwb[d + 0] = w.x; lwb[d + 2] = w.y;
        lwb[d + 4] = w.z; lwb[d + 6] = w.w;
      }
    }

    for (int k = 0; k < K; ++k) {
      __syncthreads();   // buf[k&1] staged; buf[(k+1)&1]'s readers done

      // Stage W_{k+1} into the alternate buffer; drains during compute.
      if (k + 1 < K) {
        const float4* Wk4 = (const float4*)(W + (size_t)(k + 1) * 4096);
        float* lwb = lds_w[(k + 1) & 1];
#pragma unroll
        for (int t = 0; t < 4; ++t) {
          const int i4 = tid + t * 256;
          const float4 w = Wk4[i4];
          const int r  = i4 >> 4;
          const int c0 = (i4 & 15) * 4;
          const int d  = (((r >> 1) * 64 + c0) << 1) + (r & 1);
          lwb[d + 0] = w.x; lwb[d + 2] = w.y;
          lwb[d + 4] = w.z; lwb[d + 6] = w.w;
        }
      }

      // Consume prefetched rb indices; prefetch for k+1.
      int i0 = ni0, i1 = ni1;
      if (k + 1 < K) {
        const int* rbn = rb_in + (size_t)(k + 1) * M;
        ni0 = (row0 < M) ? rbn[row0] : n_in;
        ni1 = (row1 < M) ? rbn[row1] : n_in;
      }

      const float s0 = (i0 < n_in) ? 1.0f : 0.0f;
      const float s1 = (i1 < n_in) ? 1.0f : 0.0f;
      if (i0 >= n_in) i0 = 0;              // clamp: keep loads in-bounds
      if (i1 >= n_in) i1 = 0;
      const float* a0p = feats + (size_t)i0 * 64 + 2 * hi;
      const float* a1p = feats + (size_t)i1 * 64 + 2 * hi;
      const float* lwb = lds_w[k & 1];

      // Software-pipelined K loop: prefetch A for kk+1 during kk's WMMAs.
      v2f ra0 = *(const v2f*)(a0p);
      v2f ra1 = *(const v2f*)(a1p);
#pragma unroll
      for (int kk = 0; kk < 16; ++kk) {
        v2f na0, na1;
        if (kk < 15) {
          na0 = *(const v2f*)(a0p + 4 * (kk + 1));
          na1 = *(const v2f*)(a1p + 4 * (kk + 1));
        }
        v2f a0; a0.x = ra0.x * s0; a0.y = ra0.y * s0;
        v2f a1; a1.x = ra1.x * s1; a1.y = ra1.y * s1;
#pragma unroll
        for (int nb = 0; nb < 4; ++nb) {
          const v2f b = *(const v2f*)&lwb[(((2 * kk + hi) * 64 +
                                            nb * 16 + l15) << 1)];
          acc[0][nb] = __builtin_amdgcn_wmma_f32_16x16x4_f32(
              false, a0, false, b, (short)0, acc[0][nb], false, false);
          acc[1][nb] = __builtin_amdgcn_wmma_f32_16x16x4_f32(
              false, a1, false, b, (short)0, acc[1][nb], false, false);
        }
        ra0 = na0; ra1 = na1;
      }
    }

    // Epilogue: bias (+ReLU), single coalesced store per element.
#pragma unroll
    for (int t = 0; t < 2; ++t) {
      const int base = (tile0 + t) * 16;
#pragma unroll
      for (int v = 0; v < 8; ++v) {
        const int r = base + v + 8 * hi;   // C/D row for this VGPR
        if (r < M) {
          float* orp = out + (size_t)r * 64 + l15;
#pragma unroll
          for (int nb = 0; nb < 4; ++nb) {
            float val = acc[t][nb][v] + bfr[nb];
            if (RELU) val = fmaxf(val, 0.0f);
            orp[nb * 16] = val;
          }
        }
      }
    }
  }
}

extern "C" void kernel_launch(void* const* d_in, const int* in_sizes, int n_in_cnt,
                              void* d_out, int out_size, void* d_ws, size_t ws_size,
                              hipStream_t stream) {
  (void)n_in_cnt; (void)out_size; (void)ws_size;

  const float* feats  = (const float*)d_in[0];
  const float* W1     = (const float*)d_in[1];
  const float* b1     = (const float*)d_in[2];
  const float* W2     = (const float*)d_in[3];
  const float* b2     = (const float*)d_in[4];
  const float* W3     = (const float*)d_in[5];
  const float* b3     = (const float*)d_in[6];
  const int*   rb1_in = (const int*)d_in[7];
  const int*   rb2_in = (const int*)d_in[9];

  const int n1 = in_sizes[0] / 64;   // input point count
  const int n2 = in_sizes[7] / 8;    // rb1 is [8, n2]

  float* buf1 = (float*)d_ws;
  float* buf2 = buf1 + (size_t)n2 * 64;
  float* outp = (float*)d_out;

  const int ntiles  = (n2 + 15) / 16;
  const int ngroups = (ntiles + 15) / 16;   // 8 waves x 2 tiles per block
  int grid = ngroups < 4096 ? ngroups : 4096;
  if (grid < 1) grid = 1;

  // layer 1: K=8, gather from feats[n1], +b1, ReLU
  sparse_conv_fused<1><<<grid, 256, 0, stream>>>(
      feats, W1, b1, rb1_in, buf1, 8, n1, n2, ngroups);
  // layer 2: K=27, gather from buf1[n2], +b2, ReLU
  sparse_conv_fused<1><<<grid, 256, 0, stream>>>(
      buf1, W2, b2, rb2_in, buf2, 27, n2, n2, ngroups);
  // layer 3: K=27, gather from buf2[n2], +b3, no ReLU
  sparse_conv_fused<0><<<grid, 256, 0, stream>>>(
      buf2, W3, b3, rb2_in, outp, 27, n2, n2, ngroups);
}